// CorrelationMapLayer_28956669510105
// MI455X (gfx1250) — compile-verified
//
#include <hip/hip_runtime.h>
#include <hip/hip_bf16.h>

typedef __attribute__((ext_vector_type(2))) float v2f;
typedef __attribute__((ext_vector_type(8))) float v8f;

#define BATCH 32
#define CH    512
#define HIN   56
#define WIN   56
#define SOUT  28
#define HW    (SOUT * SOUT)     // 784
#define KNN   100
#define KPAD  112               // 7 tiles of 16
#define KT    (KPAD / 16)       // 7
#define NT    (HW / 16)         // 49
#define SCALEF (55.0f / 27.0f)  // align_corners scale for 56 -> 28
#define CBSTR (HW + 4)          // padded LDS row stride

// ---------------------------------------------------------------------------
// Kernel 1: bilinear (align_corners) downsample feature1: [B,C,56,56]->[B,C,784]
// One block per (b,c) map; coalesced reads along x, contiguous writes.
// ---------------------------------------------------------------------------
__global__ void k_down_f1(const float* __restrict__ in, float* __restrict__ out) {
    const int map = blockIdx.x;                       // b*CH + c
    const float* __restrict__ src = in + (size_t)map * (HIN * WIN);
    float* __restrict__ dst = out + (size_t)map * HW;
    for (int idx = threadIdx.x; idx < HW; idx += blockDim.x) {
        int oy = idx / SOUT;
        int ox = idx - oy * SOUT;
        float cy = oy * SCALEF, cx = ox * SCALEF;
        int y0 = (int)cy, x0 = (int)cx;
        float fy = cy - (float)y0, fx = cx - (float)x0;
        int y1 = min(y0 + 1, HIN - 1), x1 = min(x0 + 1, WIN - 1);
        float v00 = src[y0 * WIN + x0], v01 = src[y0 * WIN + x1];
        float v10 = src[y1 * WIN + x0], v11 = src[y1 * WIN + x1];
        dst[idx] = (1.0f - fy) * ((1.0f - fx) * v00 + fx * v01) +
                   fy * ((1.0f - fx) * v10 + fx * v11);
    }
}

// ---------------------------------------------------------------------------
// Kernel 2: gathered downsample of feature2 at the 100 knn positions only.
// Output layout [B][KPAD][C] (c contiguous) so GEMM A-fragments are b64 loads.
// Rows k in [100,112) zero-filled so the padded GEMM rows are benign.
// ---------------------------------------------------------------------------
__global__ void k_gather_f2(const float* __restrict__ in,
                            const long long* __restrict__ knn,
                            float* __restrict__ out) {
    const int map = blockIdx.x;                       // b*CH + c
    const int b = map / CH;
    const int c = map - b * CH;
    const float* __restrict__ src = in + (size_t)map * (HIN * WIN);
    const int k = threadIdx.x;
    if (k >= KPAD) return;
    float val = 0.0f;
    if (k < KNN) {
        int x = (int)knn[2 * k];                      // knn[:,0] -> x
        int y = (int)knn[2 * k + 1];                  // knn[:,1] -> y
        float cy = y * SCALEF, cx = x * SCALEF;
        int y0 = (int)cy, x0 = (int)cx;
        float fy = cy - (float)y0, fx = cx - (float)x0;
        int y1 = min(y0 + 1, HIN - 1), x1 = min(x0 + 1, WIN - 1);
        float v00 = src[y0 * WIN + x0], v01 = src[y0 * WIN + x1];
        float v10 = src[y1 * WIN + x0], v11 = src[y1 * WIN + x1];
        val = (1.0f - fy) * ((1.0f - fx) * v00 + fx * v01) +
              fy * ((1.0f - fx) * v10 + fx * v11);
    }
    out[((size_t)b * KPAD + k) * CH + c] = val;
}

// ---------------------------------------------------------------------------
// Kernel 3: fused fp32 WMMA GEMM + relu + exp-sum normalization.
// Grid: (KT, BATCH). Block: 256 threads = 8 waves.
// Per block: corr[16 x 784] = f2_sel_tile[16 x 512] * f1d[b][512 x 784],
// then per-row denom = sum(exp(corr)), store corr/denom*10.
// ---------------------------------------------------------------------------
__global__ void k_corr_wmma(const float* __restrict__ f1d,
                            const float* __restrict__ f2sel,
                            float* __restrict__ out) {
    extern __shared__ float smem[];
    float* At      = smem;                    // [16][CH]        32 KB
    float* corrbuf = At + 16 * CH;            // [16][CBSTR]   ~50 KB
    float* red     = corrbuf + 16 * CBSTR;    // [256]
    float* dnm     = red + 256;               // [16]

    const int ktile = blockIdx.x;
    const int b     = blockIdx.y;
    const int tid   = threadIdx.x;
    const int lane  = tid & 31;
    const int wave  = tid >> 5;
    const int ln15  = lane & 15;              // A row (M) == B column (N)
    const int kgrp  = lane >> 4;              // K sub-group select

    // Stage A-tile (16 k-rows x 512 c) into LDS, fully coalesced.
    const float* __restrict__ Asrc = f2sel + ((size_t)b * KPAD + ktile * 16) * CH;
    for (int i = tid; i < 16 * CH; i += 256) At[i] = Asrc[i];
    __syncthreads();

    const float* __restrict__ Bb = f1d + (size_t)b * CH * HW;

    // Each wave owns N-tiles j, j+8, ... of the 49 tiles (784 hw columns).
    for (int j = wave; j < NT; j += 8) {
        const int hw0 = j * 16;
        v8f acc = {};
        const float* arow = At + ln15 * CH + 2 * kgrp;            // LDS, 8B aligned
        const float* bcol = Bb + (size_t)(2 * kgrp) * HW + hw0 + ln15;
        #pragma unroll 4
        for (int c0 = 0; c0 < CH; c0 += 4) {
            v2f a = *(const v2f*)(arow + c0);                     // ds_load_b64
            v2f bv;
            bv.x = bcol[(size_t)c0 * HW];
            bv.y = bcol[(size_t)c0 * HW + HW];
            acc = __builtin_amdgcn_wmma_f32_16x16x4_f32(
                /*neg_a=*/false, a, /*neg_b=*/false, bv,
                /*c_mod=*/(short)0, acc, /*reuse_a=*/false, /*reuse_b=*/false);
        }
        const int colbase = hw0 + ln15;
        #pragma unroll
        for (int r = 0; r < 8; ++r) {
            int row = r + 8 * kgrp;                               // D layout
            corrbuf[row * CBSTR + colbase] = fmaxf(acc[r], 0.0f); // relu
        }
    }
    __syncthreads();

    // Per-row sum of exp over 784 columns: 16 partial lanes per row.
    {
        const int row = tid >> 4, sub = tid & 15;
        float s = 0.0f;
        for (int col = sub; col < HW; col += 16)
            s += __expf(corrbuf[row * CBSTR + col]);
        red[tid] = s;
    }
    __syncthreads();
    if ((tid & 15) == 0) {
        float s = 0.0f;
        #pragma unroll
        for (int i = 0; i < 16; ++i) s += red[tid + i];
        dnm[tid >> 4] = 10.0f / s;                                // fold the *10
    }
    __syncthreads();

    // Coalesced normalized store; skip padded k rows >= 100.
    for (int idx = tid; idx < 16 * HW; idx += 256) {
        int row = idx / HW;
        int col = idx - row * HW;
        int kg = ktile * 16 + row;
        if (kg < KNN)
            out[((size_t)b * KNN + kg) * HW + col] = corrbuf[row * CBSTR + col] * dnm[row];
    }
}

// ---------------------------------------------------------------------------
extern "C" void kernel_launch(void* const* d_in, const int* in_sizes, int n_in,
                              void* d_out, int out_size, void* d_ws, size_t ws_size,
                              hipStream_t stream) {
    const float*     feature1 = (const float*)d_in[0];
    const float*     feature2 = (const float*)d_in[1];
    const long long* knn      = (const long long*)d_in[2];   // int64 [100][2]
    float* out = (float*)d_out;

    float* f1d   = (float*)d_ws;                              // [B][C][784]  ~51.4 MB
    float* f2sel = f1d + (size_t)BATCH * CH * HW;             // [B][112][C]  ~7.3  MB

    k_down_f1<<<BATCH * CH, 256, 0, stream>>>(feature1, f1d);
    k_gather_f2<<<BATCH * CH, 128, 0, stream>>>(feature2, knn, f2sel);

    const size_t smem_bytes = (size_t)(16 * CH + 16 * CBSTR + 256 + 16) * sizeof(float);
    k_corr_wmma<<<dim3(KT, BATCH), 256, smem_bytes, stream>>>(f1d, f2sel, out);
}